// DPConv_55293408969260
// MI455X (gfx1250) — compile-verified
//
#include <hip/hip_runtime.h>

typedef __attribute__((ext_vector_type(2))) float v2f;
typedef __attribute__((ext_vector_type(8))) float v8f;

typedef __attribute__((address_space(1))) int glb_i;   // async builtin wants non-const int*
typedef __attribute__((address_space(3))) int lds_i;

namespace {
constexpr int H = 128;
constexpr int W = 128;
constexpr int IMG = H * W;          // 16384
constexpr int TILES_PER_IMG = 64;   // 8x8 tiles of 16x16
constexpr int TS = 26;              // LDS row stride (floats): 104B -> b64-aligned pairs, conflict-free
}

#if __has_builtin(__builtin_amdgcn_global_load_async_to_lds_b32)
#define HAVE_ASYNC_LDS 1
#else
#define HAVE_ASYNC_LDS 0
#endif

static __device__ __forceinline__ void wait_async0() {
#if __has_builtin(__builtin_amdgcn_s_wait_asynccnt)
  __builtin_amdgcn_s_wait_asynccnt(0);
#else
  asm volatile("s_wait_asynccnt 0" ::: "memory");
#endif
}

// One wave (32 lanes) computes one 16x16 output tile.
// Stage:   24x24 clamped input halo -> LDS via GLOBAL_LOAD_ASYNC_TO_LDS_B32 (ASYNCcnt).
// Pass 2:  vertical 4-tap sums from LDS (ds_load_b64 column pairs) into A fragments,
//          constant banded horizontal filter as B fragments,
//          out(16x16) = V(16x24) x F(24x16) via 6x V_WMMA_F32_16X16X4_F32.
__global__ __launch_bounds__(256) void dpconv_wmma(const float* __restrict__ x,
                                                   float* __restrict__ out,
                                                   int n_tiles) {
  __shared__ float lds_tiles[8][24 * TS];

  const int lane = threadIdx.x & 31;
  const int wave = threadIdx.x >> 5;
  const int tile = blockIdx.x * 8 + wave;
  if (tile >= n_tiles) return;           // wave-uniform guard, EXEC stays full

  const int img = tile >> 6;
  const int th  = (tile >> 3) & 7;
  const int tw  = tile & 7;
  const int h0 = th * 16;
  const int w0 = tw * 16;

  const int sub  = lane & 15;
  const int half = lane >> 4;

  const float* __restrict__ xi = x + (size_t)img * IMG;
  float* __restrict__ oi = out + (size_t)img * IMG;
  float* myTile = &lds_tiles[wave][0];

  // ---- stage 24x24 clamped halo [h0-4, h0+19] x [w0-4, w0+19] into LDS ----
  {
    int r = (lane >= 24) ? 1 : 0;
    int c = lane - 24 * r;
#pragma unroll
    for (int it = 0; it < 18; ++it) {    // 18*32 = 576 = 24*24 elements
      const int gr = min(max(h0 - 4 + r, 0), H - 1);
      const int gc = min(max(w0 - 4 + c, 0), W - 1);
      const float* gp = xi + (gr * W + gc);
      float* lp = myTile + (r * TS + c);
#if HAVE_ASYNC_LDS
      __builtin_amdgcn_global_load_async_to_lds_b32((glb_i*)gp, (lds_i*)lp, 0, 0);
#else
      *lp = *gp;
#endif
      // advance linear element index by 32 = one full row (24) + 8
      c += 8;
      const int k = (c >= 24) ? 1 : 0;   // if-converted to cndmask (no EXEC change)
      c -= 24 * k;
      r += 1 + k;
    }
  }
#if HAVE_ASYNC_LDS
  wait_async0();                          // tile is wave-private: no barrier needed
#else
  __syncthreads();                        // fallback: cross-lane LDS visibility
#endif
  asm volatile("" ::: "memory");

  // ---- vertical tap rows for this lane's output row h = h0 + sub ----
  // LDS rows already hold clamped data, so tile-relative taps need no clamping:
  // pair A: rows sub+mh, sub+mh+1 (weight fva); pair B: rows +4, +5 (weight fvb)
  const int h  = h0 + sub;
  const int mh = h & 3;
  const int rt0 = sub + mh;               // <= 18
  const float fva = (h <= H - 5) ? 1.0f : 0.0f;
  const float fvb = (h >= 4)     ? 1.0f : 0.0f;
  const float rscale = 1.0f / (2.0f * (fva + fvb));

  // ---- horizontal banded-filter params for this lane's column w = w0 + sub ----
  const int w  = w0 + sub;
  const int mw = w & 3;
  const int t0 = w + mw - 4;
  const int t1 = w + mw;
  const int vw0 = (w <= W - 5) ? 1 : 0;
  const int vw1 = (w >= 4) ? 1 : 0;
  const float wscale = 1.0f / (2.0f * (float)(vw0 + vw1));

  const int cbase = w0 - 4;               // absolute column of tile col 0

  v8f acc = {};                           // C/D accumulator (16x16 f32)

#pragma unroll
  for (int q = 0; q < 6; ++q) {
    // A-matrix f32 16x4 layout: lane M = sub, K = j + 2*half -> tile col pair:
    const int ctp = 4 * q + 2 * half;     // even -> 8B-aligned with TS=26
    const float2 p0 = *(const float2*)(myTile + (rt0 + 0) * TS + ctp);
    const float2 p1 = *(const float2*)(myTile + (rt0 + 1) * TS + ctp);
    const float2 p2 = *(const float2*)(myTile + (rt0 + 4) * TS + ctp);
    const float2 p3 = *(const float2*)(myTile + (rt0 + 5) * TS + ctp);
    v2f a, b;
    a[0] = (fva * (p0.x + p1.x) + fvb * (p2.x + p3.x)) * rscale;
    a[1] = (fva * (p0.y + p1.y) + fvb * (p2.y + p3.y)) * rscale;
#pragma unroll
    for (int j = 0; j < 2; ++j) {
      // B-matrix f32 4x16 layout: lane N = sub, K = j + 2*half, row = abs col c
      const int c = cbase + ctp + j;
      const unsigned d0 = (unsigned)(c - t0);
      const unsigned d1 = (unsigned)(c - t1);
      const int hit = (vw0 & (d0 <= 1u)) | (vw1 & (d1 <= 1u));
      b[j] = hit ? wscale : 0.0f;
    }
    // D = A x B + C  (accumulate across the 6 K-chunks covering 24 columns)
    acc = __builtin_amdgcn_wmma_f32_16x16x4_f32(
        /*neg_a=*/false, a, /*neg_b=*/false, b,
        /*c_mod=*/(short)0, acc, /*reuse_a=*/false, /*reuse_b=*/false);
  }

  // ---- store D: lane column = w0+sub ; VGPR j -> row h0 + j + 8*half ----
#pragma unroll
  for (int j = 0; j < 8; ++j) {
    const int row = h0 + j + 8 * half;
    oi[row * W + (w0 + sub)] = acc[j];
  }
}

extern "C" void kernel_launch(void* const* d_in, const int* in_sizes, int n_in,
                              void* d_out, int out_size, void* d_ws, size_t ws_size,
                              hipStream_t stream) {
  (void)n_in; (void)out_size; (void)d_ws; (void)ws_size;
  const float* x = (const float*)d_in[0];
  float* out = (float*)d_out;

  const int imgs   = in_sizes[0] / IMG;        // N*C = 512
  const int tiles  = imgs * TILES_PER_IMG;     // 32768
  const int blocks = (tiles + 7) / 8;          // 8 waves (tiles) per 256-thread block

  dpconv_wmma<<<blocks, 256, 0, stream>>>(x, out, tiles);
}